// KNNImputer_44770739093702
// MI455X (gfx1250) — compile-verified
//
#include <hip/hip_runtime.h>

#define KSEL 5
#define ITERS 3

#if defined(__has_builtin)
#if __has_builtin(__builtin_amdgcn_global_load_async_to_lds_b32)
#define HAVE_ASYNC_LDS 1
#endif
#if __has_builtin(__builtin_amdgcn_s_wait_asynccnt)
#define HAVE_WAIT_ASYNC 1
#endif
#endif

typedef __attribute__((address_space(1))) void* as1_vp;
typedef __attribute__((address_space(3))) void* as3_vp;
typedef __attribute__((address_space(1))) int*  as1_i32p;
typedef __attribute__((address_space(3))) int*  as3_i32p;

// ---------------------------------------------------------------------------
// Kernel 1: out = X  (vectorized B128 copy; missing slots overwritten later)
// ---------------------------------------------------------------------------
__global__ void KNNImputer_copy_kernel(const float* __restrict__ X,
                                       float* __restrict__ out, int n4, int n) {
    int i = blockIdx.x * blockDim.x + threadIdx.x;
    if (i < n4) {
        reinterpret_cast<float4*>(out)[i] =
            reinterpret_cast<const float4*>(X)[i];
    }
    if (i == 0) {  // scalar tail (none for n = 32768, kept for generality)
        for (int k = n4 * 4; k < n; ++k) out[k] = X[k];
    }
}

// ---------------------------------------------------------------------------
// Kernel 2: single workgroup (32 wave32s on one WGP).
//  - async-gather obs_vals into LDS (118 KB of the 320 KB WGP LDS)
//  - 3 iterations of exact top-5 selection via u64 (dist,idx) keys:
//    wave32 shuffle min-reduce  ->  ds_min_u64 LDS atomic  ->  broadcast
//  - scatter the converged scalar to the missing positions
// ---------------------------------------------------------------------------
__global__ __launch_bounds__(1024) void KNNImputer_impute_kernel(
    const float* __restrict__ X, const int* __restrict__ miss_idx, int n_miss,
    const int* __restrict__ obs_idx, int n_obs, float* __restrict__ out) {
    extern __shared__ float smem[];  // n_obs gathered observed values
    __shared__ unsigned long long s_minkey;

    const int tid = threadIdx.x;
    const int nt  = blockDim.x;

    // ---- stage gathered observed values into LDS --------------------------
    for (int j = tid; j < n_obs; j += nt) {
        const float* gp = X + obs_idx[j];
#if HAVE_ASYNC_LDS
        // per-lane async gather: copies 4 bytes from global address gp to smem[j] in LDS  (tracked by ASYNCcnt)
        __builtin_amdgcn_global_load_async_to_lds_b32(
            (as1_i32p)(as1_vp)(void*)(float*)gp,
            (as3_i32p)(as3_vp)(void*)&smem[j], 0, 0);
#else
        smem[j] = *gp;
#endif
    }
#if HAVE_ASYNC_LDS
#if HAVE_WAIT_ASYNC
    __builtin_amdgcn_s_wait_asynccnt(0);
#else
    asm volatile("s_wait_asynccnt 0" ::: "memory");
#endif
#endif
    __syncthreads();

    // ---- 3 scalar fixed-point iterations ----------------------------------
    float c = 0.0f;  // all missing entries share this value (zero init)
    for (int it = 0; it < ITERS; ++it) {
        unsigned long long prev = 0ull;  // ekeys are >= 1, strictly increasing
        float sum = 0.0f;
        for (int r = 0; r < KSEL; ++r) {
            if (tid == 0) s_minkey = ~0ull;
            __syncthreads();

            // thread-local min over strided chunk, excluding already-picked keys
            unsigned long long local = ~0ull;
            for (int j = tid; j < n_obs; j += nt) {
                float v    = smem[j];
                float diff = c - v;
                float d    = diff * diff;  // matches (m - obs)**2 in f32
                unsigned long long ekey =
                    ((((unsigned long long)__float_as_uint(d)) << 32) |
                     (unsigned long long)(unsigned)j) + 1ull;
                if (ekey > prev && ekey < local) local = ekey;
            }
            // wave32 butterfly min-reduce
            for (int off = 16; off > 0; off >>= 1) {
                unsigned long long o = __shfl_xor(local, off, 32);
                if (o < local) local = o;
            }
            // cross-wave reduce: one ds_min_u64 per wave
            if ((tid & 31) == 0) atomicMin(&s_minkey, local);
            __syncthreads();

            unsigned long long w = s_minkey;  // uniform broadcast
            prev = w;
            unsigned widx = (unsigned)((w - 1ull) & 0xFFFFFFFFull);
            sum += smem[widx];  // accumulate in ascending-distance order
            __syncthreads();    // protect s_minkey re-init next round
        }
        c = sum / 5.0f;  // f32 mean, same order/precision as jnp.mean
    }

    // ---- scatter converged scalar to missing positions --------------------
    for (int i = tid; i < n_miss; i += nt) out[miss_idx[i]] = c;
}

// ---------------------------------------------------------------------------
extern "C" void kernel_launch(void* const* d_in, const int* in_sizes, int n_in,
                              void* d_out, int out_size, void* d_ws,
                              size_t ws_size, hipStream_t stream) {
    const float* X        = (const float*)d_in[0];
    const int*   miss_idx = (const int*)d_in[1];
    const int*   obs_idx  = (const int*)d_in[2];
    float*       out      = (float*)d_out;

    const int n      = in_sizes[0];  // 32768
    const int n_miss = in_sizes[1];  // 3276
    const int n_obs  = in_sizes[2];  // 29492

    const int n4 = n / 4;
    KNNImputer_copy_kernel<<<(n4 + 255) / 256, 256, 0, stream>>>(X, out, n4, n);

    const size_t shmem = (size_t)n_obs * sizeof(float);  // ~118 KB < 320 KB WGP LDS
    KNNImputer_impute_kernel<<<1, 1024, shmem, stream>>>(X, miss_idx, n_miss,
                                                         obs_idx, n_obs, out);
}